// ToHeteroLinear_53197464928985
// MI455X (gfx1250) — compile-verified
//
#include <hip/hip_runtime.h>
#include <stdint.h>

typedef float v2f __attribute__((ext_vector_type(2)));
typedef float v8f __attribute__((ext_vector_type(8)));
typedef uint32_t v4u __attribute__((ext_vector_type(4)));
typedef uint32_t v8u __attribute__((ext_vector_type(8)));

// Problem constants (from reference)
constexpr int KC = 512;   // in_channels
constexpr int OC = 512;   // out_channels

// Tiling
constexpr int BR  = 64;       // rows per block (A tile rows)
constexpr int BC  = 128;      // cols per col-tile
constexpr int BK  = 64;       // k-chunk staged for B
constexpr int NCT = OC / BC;  // 4 col tiles
constexpr int NKC = KC / BK;  // 8 k chunks
constexpr int THREADS = 256;  // 8 waves (wave32)

constexpr int LDS_A_FLOATS = BR * KC;   // 32768 floats = 128 KB
constexpr int LDS_B_FLOATS = BK * BC;   // 8192 floats  =  32 KB per buffer

// ---- gfx1250 async global->LDS bulk copy (ASYNCcnt) for the A tile ----
__device__ __forceinline__ void async_copy_b128(const float* g, float* l) {
  const uint32_t ldsaddr = (uint32_t)(uintptr_t)l;
  asm volatile("global_load_async_to_lds_b128 %0, %1, off"
               :: "v"(ldsaddr), "v"((uint64_t)(uintptr_t)g)
               : "memory");
}
__device__ __forceinline__ void wait_async_0() {
  asm volatile("s_wait_asynccnt 0x0" ::: "memory");
}

// ---- gfx1250 Tensor Data Mover: one-instruction 2D tile DMA (TENSORcnt) ----
// Loads a [BK rows] x [BC cols] f32 tile from a row-major [.., OC] tensor into LDS.
__device__ __forceinline__ void tdm_load_b_tile(uint32_t lds_byte_addr, uint64_t gaddr) {
  v4u g0;
  g0.x = 1u;                                                  // count=1, user mode
  g0.y = __builtin_amdgcn_readfirstlane(lds_byte_addr);       // lds_addr [63:32]
  g0.z = __builtin_amdgcn_readfirstlane((uint32_t)gaddr);     // global_addr lo
  g0.w = (__builtin_amdgcn_readfirstlane((uint32_t)(gaddr >> 32)) & 0x01FFFFFFu)
       | 0x80000000u;                                         // addr hi | type=2

  v8u g1;
  g1[0] = 0x00020000u;            // workgroup_mask=0, data_size=2 (4 bytes)
  g1[1] = (uint32_t)OC << 16;     // tensor_dim0 = 512 (bits 79:48, low half)
  g1[2] = (uint32_t)KC << 16;     // tensor_dim0 hi = 0 | tensor_dim1 = 512 (low half)
  g1[3] = (uint32_t)BC << 16;     // tensor_dim1 hi = 0 | tile_dim0 = 128
  g1[4] = (uint32_t)BK;           // tile_dim1 = 64 | tile_dim2 = 0
  g1[5] = (uint32_t)OC;           // tensor_dim0_stride = 512 (low 32)
  g1[6] = 0u;                     // stride hi | tensor_dim1_stride lo
  g1[7] = 0u;                     // tensor_dim1_stride hi

  asm volatile("tensor_load_to_lds %0, %1" :: "s"(g0), "s"(g1) : "memory");
}
__device__ __forceinline__ void wait_tensor_1() {   // allow the prefetched tile in flight
  asm volatile("s_wait_tensorcnt 0x1" ::: "memory");
}
__device__ __forceinline__ void wait_tensor_0() {
  asm volatile("s_wait_tensorcnt 0x0" ::: "memory");
}

__global__ __launch_bounds__(THREADS)
void hetero_linear_wmma(const float* __restrict__ x,
                        const int*   __restrict__ type_vec,
                        const float* __restrict__ weight,   // [T][KC][OC]
                        const float* __restrict__ bias,     // [T][OC]
                        float* __restrict__ out) {          // [N][OC]
  extern __shared__ float lds[];
  float* lds_a = lds;                    // [BR][KC] row-major
  float* lds_b = lds + LDS_A_FLOATS;     // 2 x [BK][BC] row-major (ping/pong)

  const int tid  = threadIdx.x;
  const int wave = tid >> 5;
  const int lane = tid & 31;
  const int m    = lane & 15;   // M (or N) within a 16x16 fragment
  const int h    = lane >> 4;   // k-half selector for A/B fragments

  // Wave-uniform scalar condition so the TDM issue lowers to s_cbranch
  // (TDM ignores EXEC; an exec-masked branch would duplicate the DMA).
  const bool is_issuer = (__builtin_amdgcn_readfirstlane(wave) == 0);

  const int blockRow = blockIdx.x * BR;

  // ---- async stage of the full A tile (x rows): 64x512 f32, HBM read once ----
  {
    const float* ga = x + (size_t)blockRow * KC;
    #pragma unroll
    for (int i = 0; i < (BR * KC / 4) / THREADS; ++i) {   // 32 b128 per lane
      const int e = (tid + i * THREADS) * 4;
      async_copy_b128(ga + e, lds_a + e);
    }
  }

  // Sorted type_vec => tile types are t0..t1 (almost always t0==t1).
  const int t0 = type_vec[blockRow];
  const int t1 = type_vec[blockRow + BR - 1];

  const int rfrag = wave & 3;          // 16-row fragment owned by this wave
  const int cgrp  = (wave >> 2) * 4;   // first of 4 col fragments owned
  const int arow  = rfrag * 16 + m;    // A-fragment row within tile (per lane)

  // Row types for the 8 C rows each lane holds (VGPR v -> row rfrag*16 + 8h + v)
  int rt[8];
  #pragma unroll
  for (int v = 0; v < 8; ++v)
    rt[v] = type_vec[blockRow + rfrag * 16 + 8 * h + v];

  const uint32_t ldsb_addr[2] = {
      (uint32_t)(uintptr_t)(lds_b),
      (uint32_t)(uintptr_t)(lds_b + LDS_B_FLOATS)};

  // All of this wave's A copies issued; reads of lds_a happen only after the
  // first barrier below, which every wave reaches only after this wait.
  wait_async_0();

  for (int ct = 0; ct < NCT; ++ct) {
    const int c0 = ct * BC;

    v8f outc[4];
    const v8f vzero = {};
    #pragma unroll
    for (int f = 0; f < 4; ++f) outc[f] = vzero;

    for (int t = t0; t <= t1; ++t) {
      const float* wt = weight + (size_t)t * KC * OC;

      // Prologue: TDM chunk 0 -> buffer 0 (previous readers finished at the
      // trailing barrier of the previous chunk loop).
      if (is_issuer)
        tdm_load_b_tile(ldsb_addr[0], (uint64_t)(uintptr_t)(wt + c0));

      v8f acc[4];
      #pragma unroll
      for (int f = 0; f < 4; ++f) acc[f] = vzero;

      for (int ic = 0; ic < NKC; ++ic) {
        const float* cur = lds_b + (ic & 1) * LDS_B_FLOATS;
        if (ic + 1 < NKC) {
          // Prefetch next W tile via TDM into the other buffer, then wait for
          // the current one (same-wave tensor ops complete in order => <=1).
          if (is_issuer)
            tdm_load_b_tile(ldsb_addr[(ic + 1) & 1],
                            (uint64_t)(uintptr_t)(wt + (size_t)(ic + 1) * BK * OC + c0));
          wait_tensor_1();
        } else {
          wait_tensor_0();
        }
        __syncthreads();   // issuer's wait => tile resident for every wave

        const int k0 = ic * BK;
        #pragma unroll
        for (int kk = 0; kk < BK; kk += 4) {
          // A 16x4 fragment: rows arow, K = k0+kk+2h, +1 (ds_load_b64)
          const v2f a = *(const v2f*)&lds_a[arow * KC + k0 + kk + 2 * h];
          #pragma unroll
          for (int f = 0; f < 4; ++f) {
            const int col = (cgrp + f) * 16 + m;
            // B 4x16 fragment: col fixed, K = kk+2h, +1 (ds_load_2addr)
            const float* bp = cur + (kk + 2 * h) * BC + col;
            v2f b;
            b.x = bp[0];
            b.y = bp[BC];
            acc[f] = __builtin_amdgcn_wmma_f32_16x16x4_f32(
                /*neg_a=*/false, a, /*neg_b=*/false, b,
                /*c_mod=*/(short)0, acc[f],
                /*reuse_a=*/false, /*reuse_b=*/false);
          }
        }
        __syncthreads();   // readers done before this buffer is refilled
      }

      // ---- merge: rows of type t take this GEMM's result + bias[t] ----
      #pragma unroll
      for (int f = 0; f < 4; ++f) {
        const int col = c0 + (cgrp + f) * 16 + m;
        const float bv = bias[(size_t)t * OC + col];
        #pragma unroll
        for (int v = 0; v < 8; ++v) {
          if (rt[v] == t) outc[f][v] = acc[f][v] + bv;
        }
      }
    }

    // ---- store col tile (coalesced across lanes) ----
    #pragma unroll
    for (int f = 0; f < 4; ++f) {
      const int col = c0 + (cgrp + f) * 16 + m;
      #pragma unroll
      for (int v = 0; v < 8; ++v) {
        const int row = blockRow + rfrag * 16 + 8 * h + v;
        out[(size_t)row * OC + col] = outc[f][v];
      }
    }
  }
}

extern "C" void kernel_launch(void* const* d_in, const int* in_sizes, int n_in,
                              void* d_out, int out_size, void* d_ws, size_t ws_size,
                              hipStream_t stream) {
  const float* x  = (const float*)d_in[0];
  const int*   tv = (const int*)d_in[1];
  const float* w  = (const float*)d_in[2];
  const float* b  = (const float*)d_in[3];
  float* out = (float*)d_out;

  const int n = in_sizes[0] / KC;          // 131072 rows
  const dim3 grid(n / BR);                 // 2048 blocks
  const size_t ldsBytes =
      (size_t)(LDS_A_FLOATS + 2 * LDS_B_FLOATS) * sizeof(float);  // 192 KB of 320 KB/WGP

  hipLaunchKernelGGL(hetero_linear_wmma, grid, dim3(THREADS), ldsBytes, stream,
                     x, tv, w, b, out);
}